// Mamba2Baseline_9234179686564
// MI455X (gfx1250) — compile-verified
//
#include <hip/hip_runtime.h>
#include <hip/hip_bf16.h>
#include <math.h>

// ---------------- problem constants (match reference) ----------------
#define BB        8
#define TT        2048
#define D_MODEL   256
#define HEADDIM   32
#define D_STATE   64
#define D_CONV    4
#define D_INNER   512               // D_MODEL * 2
#define NHEADS    16                // D_INNER / HEADDIM
#define CONV_DIM  640               // D_INNER + 2*D_STATE
#define D_IN_PROJ 1168              // 2*D_INNER + 2*D_STATE + NHEADS
#define BT        (BB * TT)         // 16384
#define EPS       1e-6f
#define GEPS      1e-5f

typedef __attribute__((ext_vector_type(2))) float v2f;
typedef __attribute__((ext_vector_type(8))) float v8f;

__device__ __forceinline__ float sigmoidf_(float x) { return 1.0f / (1.0f + __expf(-x)); }
__device__ __forceinline__ float siluf_(float x)    { return x * sigmoidf_(x); }

// ---------------------------------------------------------------------
// Kernel 1: RMSNorm over D_MODEL. One block (256 thr) per token row.
// ---------------------------------------------------------------------
__global__ void k_rmsnorm256(const float* __restrict__ x,
                             const float* __restrict__ w,
                             float* __restrict__ out)
{
    __shared__ float red[256];
    const int row = blockIdx.x;
    const int tid = threadIdx.x;
    float v = x[row * D_MODEL + tid];
    red[tid] = v * v;
    __syncthreads();
    #pragma unroll
    for (int s = 128; s > 0; s >>= 1) {
        if (tid < s) red[tid] += red[tid + s];
        __syncthreads();
    }
    float inv = rsqrtf(red[0] * (1.0f / D_MODEL) + EPS);
    out[row * D_MODEL + tid] = v * inv * w[tid];
}

// ---------------------------------------------------------------------
// WMMA fp32 GEMM:  out[M][N] = A[M][K] * W[N][K]^T  (+ optional residual)
// One wave32 per (16 x 64) output strip: 1 M-tile x 4 N-tiles. The A
// fragment is loaded once per K-step and fed to 4 WMMAs. The K loop is
// software-pipelined TWO groups deep: while the WMMAs for K-group g
// issue, the loads for group g+2 are launched, so every consumed
// fragment was loaded 8 WMMAs (plus addressing ALU) earlier and the
// s_wait_loadcnt before a group resolves to already-landed data.
//
// A fragment (16x4):  lane l holds A[mt*16 + (l&15)][k + 2*(l>>4) + j], j=0,1
// B fragment (4x16):  lane l holds W[nt*16 + (l&15)][k + 2*(l>>4) + j], j=0,1
// D (16x16 f32):      VGPR j: lanes 0-15 -> row j, lanes 16-31 -> row j+8
// ---------------------------------------------------------------------
#define NTW 4   // N-tiles per wave

__global__ void k_wmma_gemm_nt(const float* __restrict__ A,
                               const float* __restrict__ W,
                               const float* __restrict__ residual,
                               float* __restrict__ out,
                               int M, int N, int K)
{
    const int wavesPerBlock = blockDim.x >> 5;
    const int waveId = (blockIdx.x * wavesPerBlock) + (threadIdx.x >> 5);
    const int ntN  = N >> 4;
    const int ntM  = M >> 4;
    const int ntN4 = (ntN + NTW - 1) / NTW;
    if (waveId >= ntM * ntN4) return;
    const int mt  = waveId / ntN4;
    const int nt0 = (waveId % ntN4) * NTW;

    const int lane = threadIdx.x & 31;
    const int l15  = lane & 15;
    const int koff = (lane >> 4) << 1;          // 0 or 2

    const float* __restrict__ arow = A + (size_t)(mt * 16 + l15) * K + koff;

    const float* wrow[NTW];
    bool valid[NTW];
    #pragma unroll
    for (int i = 0; i < NTW; ++i) {
        int nt   = nt0 + i;
        valid[i] = (nt < ntN);
        wrow[i]  = W + (size_t)((valid[i] ? nt : 0) * 16 + l15) * K + koff;
    }

    v8f acc[NTW];
    #pragma unroll
    for (int i = 0; i < NTW; ++i) acc[i] = (v8f){};

    // ---- prologue: load K-groups 0 and 1 ----
    v2f a0 = *(const v2f*)(arow);
    v2f a1 = *(const v2f*)(arow + 4);
    v2f b0[NTW], b1[NTW];
    #pragma unroll
    for (int i = 0; i < NTW; ++i) {
        b0[i] = *(const v2f*)(wrow[i]);
        b1[i] = *(const v2f*)(wrow[i] + 4);
    }

    const int G = K >> 2;                       // number of K-groups
    #pragma unroll 2
    for (int g = 0; g < G - 2; ++g) {
        const int kpf = (g + 2) << 2;           // prefetch group g+2
        v2f an = *(const v2f*)(arow + kpf);
        v2f bn[NTW];
        #pragma unroll
        for (int i = 0; i < NTW; ++i) bn[i] = *(const v2f*)(wrow[i] + kpf);

        #pragma unroll
        for (int i = 0; i < NTW; ++i)
            acc[i] = __builtin_amdgcn_wmma_f32_16x16x4_f32(
                         false, a0, false, b0[i], (short)0, acc[i], false, false);

        a0 = a1; a1 = an;
        #pragma unroll
        for (int i = 0; i < NTW; ++i) { b0[i] = b1[i]; b1[i] = bn[i]; }
    }
    // ---- epilogue: drain the last two groups ----
    #pragma unroll
    for (int i = 0; i < NTW; ++i)
        acc[i] = __builtin_amdgcn_wmma_f32_16x16x4_f32(
                     false, a0, false, b0[i], (short)0, acc[i], false, false);
    #pragma unroll
    for (int i = 0; i < NTW; ++i)
        acc[i] = __builtin_amdgcn_wmma_f32_16x16x4_f32(
                     false, a1, false, b1[i], (short)0, acc[i], false, false);

    const int rbase = mt * 16 + ((lane >> 4) << 3);   // + j gives the row
    #pragma unroll
    for (int i = 0; i < NTW; ++i) {
        if (!valid[i]) continue;
        const int col = (nt0 + i) * 16 + l15;
        #pragma unroll
        for (int j = 0; j < 8; ++j) {
            size_t idx = (size_t)(rbase + j) * N + col;
            float v = acc[i][j];
            if (residual) v += residual[idx];
            out[idx] = v;
        }
    }
}

// ---------------------------------------------------------------------
// Kernel 3: depthwise causal conv1d (width 4) + bias + SiLU over xBC.
// Input = zxbcdt rows (stride D_IN_PROJ), columns [D_INNER, D_INNER+CONV_DIM)
// Output = compact (BT x CONV_DIM) buffer.
// ---------------------------------------------------------------------
__global__ void k_conv_silu(const float* __restrict__ zx,
                            const float* __restrict__ cw,
                            const float* __restrict__ cb,
                            float* __restrict__ out)
{
    const int idx = blockIdx.x * blockDim.x + threadIdx.x;
    const int total = BT * CONV_DIM;
    if (idx >= total) return;
    const int c  = idx % CONV_DIM;
    const int bt = idx / CONV_DIM;
    const int t  = bt % TT;
    const int btBase = bt - t;

    float accv = cb[c];
    #pragma unroll
    for (int i = 0; i < D_CONV; ++i) {
        int ts = t - (D_CONV - 1) + i;
        if (ts >= 0)
            accv += zx[(size_t)(btBase + ts) * D_IN_PROJ + D_INNER + c] * cw[c * D_CONV + i];
    }
    out[(size_t)bt * CONV_DIM + c] = siluf_(accv);
}

// ---------------------------------------------------------------------
// Kernel 4: selective-state scan. One block = 32 lanes = one (b,h) pair.
// Lane p owns state row h[p][0..63] in VGPRs. B/C broadcast via LDS.
// ---------------------------------------------------------------------
__global__ void k_ssm_scan(const float* __restrict__ zx,      // for dt columns
                           const float* __restrict__ xbc,     // conv output (BT x 640)
                           const float* __restrict__ dt_bias,
                           const float* __restrict__ A_log,
                           const float* __restrict__ Dvec,
                           float* __restrict__ y)             // (BT x 512), d = h*32+p
{
    const int b = blockIdx.x / NHEADS;
    const int h = blockIdx.x % NHEADS;
    const int p = threadIdx.x;                 // 0..31

    const float Ah  = -__expf(A_log[h]);
    const float Dh  = Dvec[h];
    const float dtb = dt_bias[h];

    float hs[D_STATE];
    #pragma unroll
    for (int n = 0; n < D_STATE; ++n) hs[n] = 0.0f;

    __shared__ __align__(16) float sBC[2 * D_STATE];   // [0..63]=B, [64..127]=C

    for (int t = 0; t < TT; ++t) {
        const size_t bt = (size_t)b * TT + t;
        const float* row = xbc + bt * CONV_DIM;

        // per-lane x value
        const float xp = row[h * HEADDIM + p];

        // stage B (cols 512..575) and C (cols 576..639) into LDS
        sBC[p]      = row[D_INNER + p];
        sBC[p + 32] = row[D_INNER + 32 + p];
        sBC[p + 64] = row[D_INNER + D_STATE + p];
        sBC[p + 96] = row[D_INNER + D_STATE + 32 + p];
        __syncthreads();

        // dt = softplus(dt_raw + bias); dA = exp(dt*A)
        float dtr = zx[bt * D_IN_PROJ + (D_IN_PROJ - NHEADS) + h] + dtb;
        float dt  = (dtr > 20.0f) ? dtr : log1pf(__expf(dtr));
        float dA  = __expf(dt * Ah);
        float coef = dt * xp;

        float accv = 0.0f;
        const float4* sB4 = (const float4*)(sBC);
        const float4* sC4 = (const float4*)(sBC + D_STATE);
        #pragma unroll
        for (int q = 0; q < D_STATE / 4; ++q) {
            float4 bb = sB4[q];
            float4 cc = sC4[q];
            int n = q * 4;
            hs[n + 0] = hs[n + 0] * dA + coef * bb.x;  accv += hs[n + 0] * cc.x;
            hs[n + 1] = hs[n + 1] * dA + coef * bb.y;  accv += hs[n + 1] * cc.y;
            hs[n + 2] = hs[n + 2] * dA + coef * bb.z;  accv += hs[n + 2] * cc.z;
            hs[n + 3] = hs[n + 3] * dA + coef * bb.w;  accv += hs[n + 3] * cc.w;
        }
        __syncthreads();

        y[bt * D_INNER + h * HEADDIM + p] = accv + Dh * xp;
    }
}

// ---------------------------------------------------------------------
// Kernel 5: gated RMSNorm: yn = rms(y * silu(z)) * gnorm_w over D_INNER.
// One block (256 thr) per token row, 2 elements per thread.
// ---------------------------------------------------------------------
__global__ void k_gated_rmsnorm(const float* __restrict__ y,
                                const float* __restrict__ zx,   // z = cols [0,512)
                                const float* __restrict__ gw,
                                float* __restrict__ out)
{
    __shared__ float red[256];
    const int row = blockIdx.x;
    const int tid = threadIdx.x;

    float g0, g1;
    {
        float z0 = zx[(size_t)row * D_IN_PROJ + tid];
        float z1 = zx[(size_t)row * D_IN_PROJ + tid + 256];
        float y0 = y[(size_t)row * D_INNER + tid];
        float y1 = y[(size_t)row * D_INNER + tid + 256];
        g0 = y0 * siluf_(z0);
        g1 = y1 * siluf_(z1);
    }
    red[tid] = g0 * g0 + g1 * g1;
    __syncthreads();
    #pragma unroll
    for (int s = 128; s > 0; s >>= 1) {
        if (tid < s) red[tid] += red[tid + s];
        __syncthreads();
    }
    float inv = rsqrtf(red[0] * (1.0f / D_INNER) + GEPS);
    out[(size_t)row * D_INNER + tid]       = g0 * inv * gw[tid];
    out[(size_t)row * D_INNER + tid + 256] = g1 * inv * gw[tid + 256];
}

// ---------------------------------------------------------------------
extern "C" void kernel_launch(void* const* d_in, const int* in_sizes, int n_in,
                              void* d_out, int out_size, void* d_ws, size_t ws_size,
                              hipStream_t stream)
{
    const float* x         = (const float*)d_in[0];  // (B,T,256)
    const float* norm_w    = (const float*)d_in[1];  // (256)
    const float* in_proj_w = (const float*)d_in[2];  // (1168,256)
    const float* conv_w    = (const float*)d_in[3];  // (640,4)
    const float* conv_b    = (const float*)d_in[4];  // (640)
    const float* dt_bias   = (const float*)d_in[5];  // (16)
    const float* A_log     = (const float*)d_in[6];  // (16)
    const float* Dvec      = (const float*)d_in[7];  // (16)
    const float* gnorm_w   = (const float*)d_in[8];  // (512)
    const float* out_proj_w= (const float*)d_in[9];  // (256,512)
    float* out = (float*)d_out;                      // (B,T,256)

    // workspace layout (floats)
    float* xn = (float*)d_ws;                  // BT*256
    float* zx = xn + (size_t)BT * D_MODEL;     // BT*1168
    float* xc = zx + (size_t)BT * D_IN_PROJ;   // BT*640
    float* ys = xc + (size_t)BT * CONV_DIM;    // BT*512
    float* yn = ys + (size_t)BT * D_INNER;     // BT*512

    // 1) RMSNorm of x
    k_rmsnorm256<<<BT, 256, 0, stream>>>(x, norm_w, xn);

    // 2) in_proj GEMM: zx = xn @ in_proj_w^T   (M=BT, N=1168, K=256)
    {
        int ntN4 = (D_IN_PROJ / 16 + NTW - 1) / NTW;
        int tiles = (BT / 16) * ntN4;
        int blocks = (tiles + 7) / 8;                    // 8 waves / block
        k_wmma_gemm_nt<<<blocks, 256, 0, stream>>>(xn, in_proj_w, nullptr, zx,
                                                   BT, D_IN_PROJ, D_MODEL);
    }

    // 3) depthwise causal conv + SiLU on xBC slice
    {
        int total = BT * CONV_DIM;
        k_conv_silu<<<(total + 255) / 256, 256, 0, stream>>>(zx, conv_w, conv_b, xc);
    }

    // 4) selective scan: one wave per (b,h)
    k_ssm_scan<<<BB * NHEADS, 32, 0, stream>>>(zx, xc, dt_bias, A_log, Dvec, ys);

    // 5) gated RMSNorm
    k_gated_rmsnorm<<<BT, 256, 0, stream>>>(ys, zx, gnorm_w, yn);

    // 6) out_proj GEMM + residual: out = x + yn @ out_proj_w^T (M=BT,N=256,K=512)
    {
        int ntN4 = (D_MODEL / 16 + NTW - 1) / NTW;
        int tiles = (BT / 16) * ntN4;
        int blocks = (tiles + 7) / 8;
        k_wmma_gemm_nt<<<blocks, 256, 0, stream>>>(yn, out_proj_w, x, out,
                                                   BT, D_MODEL, D_INNER);
    }
}